// disparity_60876866453981
// MI455X (gfx1250) — compile-verified
//
#include <hip/hip_runtime.h>
#include <hip/hip_bf16.h>
#include <math.h>

// ---------------------------------------------------------------------------
// disparity loss on MI455X (gfx1250)
//
// outputs: [B=16, C=512, H=128, W=128] f32  (512 MB -> HBM-bound, ~22us floor)
// masks:   [B=16, H=128, W=128] int32, values 0..8
//
// sums[b,p,c] = sum_{hw} (mask[b,hw]==p+1) * outputs[b,c,hw]  via
// V_WMMA_F32_16X16X4_F32:  A = onehot (M=class rows 0..15, K=4 pixels),
//                          B = outputs (K=4 pixels, N=16 channels).
// Full f32 precision, no conversion VALU, 1024 MACs/instr.
// ---------------------------------------------------------------------------

typedef __attribute__((ext_vector_type(2))) float v2f;
typedef __attribute__((ext_vector_type(8))) float v8f;

#define NB   16
#define NC   512
#define HW   16384   // 128*128
#define NP   8       // foreground classes 1..8

// ---------------------------------------------------------------------------
// workspace zeroing (harness poisons d_ws; atomics below need zeros each call)
// ---------------------------------------------------------------------------
__global__ void dl_zero_ws(float* __restrict__ sums, int* __restrict__ counts) {
  int i = blockIdx.x * blockDim.x + threadIdx.x;          // grid covers 65536
  if (i < NB * NP * NC) sums[i] = 0.0f;
  if (i < NB * NP)      counts[i] = 0;
}

// ---------------------------------------------------------------------------
// counts[b,p] = #pixels of class p+1 in image b (tiny: 1MB of masks)
// ---------------------------------------------------------------------------
__global__ __launch_bounds__(256)
void dl_hist(const int* __restrict__ masks, int* __restrict__ counts) {
  int i = blockIdx.x * 256 + threadIdx.x;                 // 0..262143
  int m = masks[i];
  int b = i >> 14;                                        // i / HW
  if (m > 0) atomicAdd(&counts[b * NP + (m - 1)], 1);
}

// ---------------------------------------------------------------------------
// Main streaming kernel: one wave owns a 16x16 (class x channel) f32 tile.
// grid = (32 channel tiles, 16 batches, 4 HW segments), block = 128 (4 waves).
// Each wave marches 1024 pixels in K=4 steps:
//   B-tile: lane (N = lane&15) reads float2 at pixel hw + klo,
//           klo = 0 for lanes 0-15, 2 for lanes 16-31
//           -> 16x4 f32 B striping (VGPR0: K=klo, VGPR1: K=klo+1).
//   A-tile: onehot(mask == (lane&15)+1) for the same two pixels
//           (rows 8..15 compare against classes 9..16 -> always 0).
//
// Loop shape: literal trip count (scalar s_cmp loop, no exec masking),
// 8 WMMA steps fully unrolled with constant byte offsets so 16 b64 loads
// can be claused/in-flight at once, two accumulators to break the
// WMMA C->C dependency chain.
// ---------------------------------------------------------------------------
__global__ __launch_bounds__(128)
void dl_seg_sums_wmma(const float* __restrict__ outputs,
                      const int*   __restrict__ masks,
                      float*       __restrict__ sums) {
  const int ct   = blockIdx.x;            // channel tile 0..31
  const int b    = blockIdx.y;            // batch 0..15
  const int seg  = blockIdx.z;            // HW segment 0..3
  const int tid  = threadIdx.x;
  const int wave = tid >> 5;              // 0..3
  const int lane = tid & 31;
  const int col  = lane & 15;             // N (channel-in-tile) == M (class row)
  const int klo  = (lane >> 4) << 1;      // 0 or 2
  const int cls  = col + 1;               // class value this row matches

  const int hw0 = seg * 4096 + wave * 1024;   // folded into bases (uniform loop)

  const float* bp = outputs
      + ((size_t)b * NC + (size_t)ct * 16 + col) * HW + hw0 + klo;
  const int* mp = masks + (size_t)b * HW + hw0 + klo;

  v8f acc0 = {};
  v8f acc1 = {};

  for (int it = 0; it < 1024; it += 32) {   // 32 pixels / outer iteration
#pragma unroll
    for (int u = 0; u < 8; ++u) {           // 8 WMMAs, constant offsets
      const int o = it + 4 * u;
      v2f bv = *(const v2f*)(bp + o);       // global_load_b64, offset imm
      int2 mm = *(const int2*)(mp + o);     // global_load_b64 (broadcast-ish)
      v2f av;
      av.x = (mm.x == cls) ? 1.0f : 0.0f;
      av.y = (mm.y == cls) ? 1.0f : 0.0f;
      if (u & 1)
        acc1 = __builtin_amdgcn_wmma_f32_16x16x4_f32(
            false, av, false, bv, (short)0, acc1, false, false);
      else
        acc0 = __builtin_amdgcn_wmma_f32_16x16x4_f32(
            false, av, false, bv, (short)0, acc0, false, false);
    }
  }

  // D layout (32-bit C/D 16x16): lanes 0-15, VGPR j <-> M=j, N=lane.
  // Rows 0..7 (classes 1..8) live in lanes 0..15; rows 8..15 are zero.
  if (lane < 16) {
    float* srow = sums + (size_t)b * NP * NC + (size_t)ct * 16 + col;
#pragma unroll
    for (int j = 0; j < NP; ++j)
      atomicAdd(srow + (size_t)j * NC, acc0[j] + acc1[j]);
  }
}

// ---------------------------------------------------------------------------
// Finalize (single block): prototypes -> pairwise distances -> 1/mean.
// mean over 8x8 matrix incl. zero diagonal: mean = 2*S/64 -> out = 32/S.
// ---------------------------------------------------------------------------
__global__ __launch_bounds__(256)
void dl_finalize(const float* __restrict__ sums,
                 const int*   __restrict__ counts,
                 float*       __restrict__ out) {
  __shared__ int   cnt_l[NB * NP];
  __shared__ float protos[NP * NC];       // 16 KB
  __shared__ float Ssum;

  const int tid = threadIdx.x;
  if (tid < NB * NP) cnt_l[tid] = counts[tid];
  if (tid == 0) Ssum = 0.0f;
  __syncthreads();

  // n_present[p] = max(#images where class present, 1)
  float npres[NP];
#pragma unroll
  for (int p = 0; p < NP; ++p) {
    int np = 0;
    for (int bb = 0; bb < NB; ++bb) np += (cnt_l[bb * NP + p] > 0) ? 1 : 0;
    npres[p] = (np > 0) ? (float)np : 1.0f;
  }

  // prototypes: thread owns 2 channels; loads of sums coalesce over c.
  for (int c = tid * 2; c < tid * 2 + 2; ++c) {
#pragma unroll
    for (int p = 0; p < NP; ++p) {
      float a = 0.0f;
      for (int bb = 0; bb < NB; ++bb) {
        int cn = cnt_l[bb * NP + p];
        if (cn > 0)
          a += sums[((size_t)bb * NP + p) * NC + c] / (float)cn;
      }
      protos[p * NC + c] = a / npres[p];
    }
  }
  __syncthreads();

  // 28 unordered pairs distributed over 8 waves.
  const int wave = tid >> 5, lane = tid & 31;
  for (int k = wave; k < 28; k += 8) {
    int p = 0, kk = k;
    while (kk >= 7 - p) { kk -= 7 - p; ++p; }
    int q = p + 1 + kk;
    float d2 = 0.0f;
    for (int c = lane; c < NC; c += 32) {
      float df = protos[p * NC + c] - protos[q * NC + c];
      d2 += df * df;
    }
    for (int off = 16; off > 0; off >>= 1)
      d2 += __shfl_down(d2, off, 32);
    if (lane == 0) {
      float d = (d2 > 0.0f) ? sqrtf(d2) : 0.0f;
      atomicAdd(&Ssum, d);                // ds_add_f32
    }
  }
  __syncthreads();
  if (tid == 0) out[0] = 32.0f / Ssum;    // 1 / mean(dist), 64 entries
}

// ---------------------------------------------------------------------------
extern "C" void kernel_launch(void* const* d_in, const int* in_sizes, int n_in,
                              void* d_out, int out_size, void* d_ws, size_t ws_size,
                              hipStream_t stream) {
  const float* outputs = (const float*)d_in[0];   // [16,512,128,128] f32
  const int*   masks   = (const int*)d_in[1];     // [16,128,128] i32
  float* out = (float*)d_out;                     // scalar f32

  float* sums   = (float*)d_ws;                               // 65536 f32
  int*   counts = (int*)((char*)d_ws + (size_t)NB * NP * NC * sizeof(float));

  dl_zero_ws<<<dim3(256), dim3(256), 0, stream>>>(sums, counts);
  dl_hist<<<dim3(1024), dim3(256), 0, stream>>>(masks, counts);
  dl_seg_sums_wmma<<<dim3(32, 16, 4), dim3(128), 0, stream>>>(outputs, masks, sums);
  dl_finalize<<<dim3(1), dim3(256), 0, stream>>>(sums, counts, out);
}